// FractionalDilationR2_10101763080232
// MI455X (gfx1250) — compile-verified
//
#include <hip/hip_runtime.h>
#include <hip/hip_bf16.h>

#define RAD     5
#define KS      11
#define TS      64
#define HALO    (TS + 2 * RAD)   // 74
#define LSTRIDE 76               // rows 16B-aligned (76*4 = 304 = 19*16)
#define NEGF    (-1.0e30f)
#define EEXP    (13.0f / 3.0f)   // 2a/(2a-1), a = 0.65
#define NUC     0.09624243f      // (2a-1)*(2a)^(-e)
#define HW      512
#define NPLANE  256              // B*C = 8*32

typedef __attribute__((ext_vector_type(2))) float v2f;
typedef __attribute__((ext_vector_type(8))) float v8f;

// ---------------------------------------------------------------------------
// Kernel A: morphological kernel table [32 x 121] via f32 WMMA.
//   g[c,p] = sum_k fp[c,k] * B_k(theta_p)   (32x6 @ 6x121, K padded to 8)
//   kern[c,p] = nu * rho_p^e * exp(-e * g[c,p])
// grid = 16 waves: blockIdx -> (mtile = channels/16, ntile = positions/16)
// ---------------------------------------------------------------------------
__global__ __launch_bounds__(32)
void fd_kern_gen(const float* __restrict__ fp, float* __restrict__ kern) {
    const int lane  = threadIdx.x;        // 0..31
    const int hi    = lane >> 4;          // which half-wave
    const int lo    = lane & 15;
    const int mtile = blockIdx.x >> 3;    // 0..1
    const int ntile = blockIdx.x & 7;     // 0..7
    const int p     = ntile * 16 + lo;    // grid position (column of B and D)

    // Circular-harmonic basis at position p (clamp p>=121; those columns unused).
    const int   pc = (p < 121) ? p : 0;
    const int   iy = pc / 11;
    const int   ix = pc - 11 * iy;
    const float yy = (float)(iy - RAD);
    const float xx = (float)(ix - RAD);
    const float th = atan2f(yy, xx);
    float s1, c1;
    sincosf(th, &s1, &c1);
    const float c2 = c1 * c1 - s1 * s1;
    const float s2 = 2.0f * c1 * s1;
    const float c3 = c2 * c1 - s2 * s1;
    const float s3 = s2 * c1 + c2 * s1;

    // A fragment: 16x4 f32, lanes 0-15 hold K0/K1, lanes 16-31 hold K2/K3.
    const float* __restrict__ fa = fp + (mtile * 16 + lo) * 6;
    v2f a1, a2, b1, b2;
    a1.x = fa[2 * hi];
    a1.y = fa[2 * hi + 1];
    a2.x = hi ? 0.0f : fa[4];   // K=4..7 (6,7 zero pad)
    a2.y = hi ? 0.0f : fa[5];
    // B fragment: 4x16 f32, row K striped across lane halves (mirrors A).
    b1.x = hi ? c2 : c1;        // K=0:cos t | K=2:cos 2t
    b1.y = hi ? s2 : s1;        // K=1:sin t | K=3:sin 2t
    b2.x = hi ? 0.0f : c3;      // K=4:cos 3t
    b2.y = hi ? 0.0f : s3;      // K=5:sin 3t

    v8f g = {};
#if defined(__has_builtin) && __has_builtin(__builtin_amdgcn_wmma_f32_16x16x4_f32)
    g = __builtin_amdgcn_wmma_f32_16x16x4_f32(false, a1, false, b1, (short)0, g, false, false);
    g = __builtin_amdgcn_wmma_f32_16x16x4_f32(false, a2, false, b2, (short)0, g, false, false);
#else
    // Scalar fallback matching the D-matrix layout below.
    const float Bk[6] = {c1, s1, c2, s2, c3, s3};
    #pragma unroll
    for (int j = 0; j < 8; ++j) {
        const float* fr = fp + (mtile * 16 + j + 8 * hi) * 6;
        float acc = 0.0f;
        #pragma unroll
        for (int k = 0; k < 6; ++k) acc += fr[k] * Bk[k];
        g[j] = acc;
    }
#endif

    // D layout: VGPR j -> M = j (lanes 0-15) / j+8 (lanes 16-31), N = lo.
    if (p < 121) {
        const float rho = sqrtf(xx * xx + yy * yy);
        const float rpe = powf(rho, EEXP);          // rho^e (0 at center)
        #pragma unroll
        for (int j = 0; j < 8; ++j) {
            const int c = mtile * 16 + j + 8 * hi;
            kern[c * 121 + p] = NUC * rpe * expf(-EEXP * g[j]);
        }
    }
}

// ---------------------------------------------------------------------------
// Kernel B: max-plus dilation. 64x64 output tile per 256-thread block,
// LDS-tiled input with radius-5 halo; each thread computes a 1x16 strip via a
// 28-float register row cache; taps paired so the compiler can form v_max3.
// Kernel taps are block-uniform -> scalar (SGPR) loads.
// ---------------------------------------------------------------------------
__global__ __launch_bounds__(256)
void fd_dilate(const float* __restrict__ in, const float* __restrict__ kern,
               float* __restrict__ out) {
    __shared__ __align__(16) float tile[HALO * LSTRIDE];

    const int plane = blockIdx.z;                 // b*32 + c
    const int chan  = plane & 31;
    const int ty0   = blockIdx.y * TS;
    const int tx0   = blockIdx.x * TS;
    const float* __restrict__ inp = in + (size_t)plane * (HW * HW);
    const float* __restrict__ kc  = kern + chan * 121;   // wave-uniform

    // --- load input tile + halo into LDS (OOB -> -1e30) ---
    const int t = threadIdx.x;
    for (int n = t; n < HALO * HALO; n += 256) {
        const int ly = n / HALO;
        const int lx = n - ly * HALO;
        const int gy = ty0 + ly - RAD;
        const int gx = tx0 + lx - RAD;
        float v = NEGF;
        if ((unsigned)gy < (unsigned)HW && (unsigned)gx < (unsigned)HW)
            v = inp[gy * HW + gx];
        tile[ly * LSTRIDE + lx] = v;
    }
    __syncthreads();

    const int r = t >> 2;   // output row in tile, 0..63
    const int s = t & 3;    // 16-wide strip, 0..3

    float acc[16];
    #pragma unroll
    for (int i = 0; i < 16; ++i) acc[i] = NEGF;

    #pragma unroll 1
    for (int dy = 0; dy < KS; ++dy) {
        // Register row cache: 28 floats (need 26), 16B-aligned float4 LDS reads.
        float rc[28];
        const float4* __restrict__ rp =
            (const float4*)&tile[(r + dy) * LSTRIDE + s * 16];
        #pragma unroll
        for (int q = 0; q < 7; ++q) {
            const float4 v = rp[q];
            rc[4 * q + 0] = v.x; rc[4 * q + 1] = v.y;
            rc[4 * q + 2] = v.z; rc[4 * q + 3] = v.w;
        }
        // Paired taps -> 2 subs + (v_max3-fusable) max chain per 2 taps.
        #pragma unroll
        for (int dx = 0; dx < KS - 1; dx += 2) {
            const float k0 = kc[dy * KS + dx];
            const float k1 = kc[dy * KS + dx + 1];
            #pragma unroll
            for (int i = 0; i < 16; ++i)
                acc[i] = fmaxf(acc[i],
                               fmaxf(rc[i + dx] - k0, rc[i + dx + 1] - k1));
        }
        {
            const float k0 = kc[dy * KS + (KS - 1)];
            #pragma unroll
            for (int i = 0; i < 16; ++i)
                acc[i] = fmaxf(acc[i], rc[i + (KS - 1)] - k0);
        }
    }

    // --- store 16 contiguous outputs as 4 float4s ---
    float4* __restrict__ op =
        (float4*)(out + (size_t)plane * (HW * HW) + (size_t)(ty0 + r) * HW
                  + tx0 + s * 16);
    #pragma unroll
    for (int q = 0; q < 4; ++q) {
        float4 v;
        v.x = acc[4 * q + 0]; v.y = acc[4 * q + 1];
        v.z = acc[4 * q + 2]; v.w = acc[4 * q + 3];
        op[q] = v;
    }
}

// ---------------------------------------------------------------------------
extern "C" void kernel_launch(void* const* d_in, const int* in_sizes, int n_in,
                              void* d_out, int out_size, void* d_ws, size_t ws_size,
                              hipStream_t stream) {
    const float* inp = (const float*)d_in[0];   // [8,32,512,512] f32
    const float* fp  = (const float*)d_in[1];   // [32,6] f32
    float* outp = (float*)d_out;                // [8,32,512,512] f32
    float* kern = (float*)d_ws;                 // [32,121] f32 scratch

    fd_kern_gen<<<dim3(16), dim3(32), 0, stream>>>(fp, kern);
    fd_dilate<<<dim3(HW / TS, HW / TS, NPLANE), dim3(256), 0, stream>>>(inp, kern, outp);
}